// GNN_28415503630843
// MI455X (gfx1250) — compile-verified
//
#include <hip/hip_runtime.h>
#include <hip/hip_bf16.h>
#include <math.h>

// Problem constants (match reference)
#define NN 50000
#define EE 1600000
#define DD 64
#define HH 2
#define HD 128            // H*D
#define NEG_SLOPE 0.2f

typedef float v2f __attribute__((ext_vector_type(2)));
typedef float v8f __attribute__((ext_vector_type(8)));

// ---------------------------------------------------------------------------
// Kernel 0: h = x @ W + b_W  using V_WMMA_F32_16X16X4_F32 (full fp32 WMMA).
// One wave computes one 16x16 output tile; K=64 -> 16 wmma steps.
// 50000 rows = 3125 tiles exactly; 128 cols = 8 tiles. 25000 waves total.
// ---------------------------------------------------------------------------
__global__ __launch_bounds__(128)
void gemm_xw(const float* __restrict__ x, const float* __restrict__ W,
             const float* __restrict__ bW, float* __restrict__ hbuf) {
  const int lane    = threadIdx.x & 31;
  const int wave    = blockIdx.x * 4 + (threadIdx.x >> 5);
  const int tileRow = wave >> 3;       // 0..3124
  const int tileCol = wave & 7;        // 0..7
  const int half    = lane >> 4;       // 0 or 1
  const int m16     = lane & 15;
  const int rowBase = tileRow * 16;
  const int colBase = tileCol * 16;

  v8f c = {};
#pragma unroll
  for (int kb = 0; kb < 16; ++kb) {
    const int k0 = kb * 4 + half * 2;  // this half-wave's K pair
    v2f a, b;
    // A 16x4 (MxK): lanes 0-15 -> K=k0,k0+1 for M=m16; lanes 16-31 -> K=k0,k0+1 (k0 = 4*kb+2)
    a.x = x[(rowBase + m16) * DD + k0 + 0];
    a.y = x[(rowBase + m16) * DD + k0 + 1];
    // B 4x16 (KxN): row K striped across lanes within a VGPR, mirrored halves
    b.x = W[(k0 + 0) * HD + colBase + m16];
    b.y = W[(k0 + 1) * HD + colBase + m16];
    c = __builtin_amdgcn_wmma_f32_16x16x4_f32(false, a, false, b, (short)0, c,
                                              false, false);
  }
  // C/D 16x16: VGPR g holds M = g (lanes 0-15) / M = g+8 (lanes 16-31), N = lane%16
  const int   col = colBase + m16;
  const float bv  = bW[col];
#pragma unroll
  for (int g = 0; g < 8; ++g) {
    const int row = rowBase + half * 8 + g;
    hbuf[row * HD + col] = c[g] + bv;
  }
}

// ---------------------------------------------------------------------------
// Kernel 1: out = bias (broadcast), m = -inf, s = 0. Re-run every launch.
// ---------------------------------------------------------------------------
__global__ __launch_bounds__(256)
void init_buffers(float* __restrict__ out, const float* __restrict__ bias,
                  float* __restrict__ m, float* __restrict__ s) {
  const int idx = blockIdx.x * 256 + threadIdx.x;
  if (idx < NN * DD) out[idx] = bias[idx & (DD - 1)];
  if (idx < NN * HH) { m[idx] = -__builtin_inff(); s[idx] = 0.f; }
}

// Float atomic-max via signed/unsigned integer ordering trick (monotone map).
__device__ inline void atomicMaxF(float* addr, float val) {
  if (val >= 0.f) atomicMax((int*)addr, __float_as_int(val));
  else            atomicMin((unsigned int*)addr, __float_as_uint(val));
}

// ---------------------------------------------------------------------------
// Kernel 2: per-edge attention logits + segment max (over row = src node).
// One wave (32 lanes) per edge; each lane owns 2 of the 64 feature dims.
// ---------------------------------------------------------------------------
__global__ __launch_bounds__(256)
void edge_alpha(const float* __restrict__ hbuf, const int* __restrict__ eidx,
                const int* __restrict__ eattr, const float* __restrict__ att,
                const float* __restrict__ bemb, float* __restrict__ alpha,
                float* __restrict__ m) {
  const int lane = threadIdx.x & 31;
  const int e    = blockIdx.x * 8 + (threadIdx.x >> 5);
  if (e >= EE) return;
  const int row = eidx[e];
  const int col = eidx[EE + e];
  const int a0  = eattr[e * 3 + 0];
  const int a1  = eattr[e * 3 + 1];
  const int a2  = eattr[e * 3 + 2];

  float p0 = 0.f, p1 = 0.f;
#pragma unroll
  for (int dd = 0; dd < 2; ++dd) {
    const int d = lane * 2 + dd;
#pragma unroll
    for (int h = 0; h < 2; ++h) {
      const int hd = h * DD + d;
      const float em = bemb[(0 * 5 + a0) * HD + hd]
                     + bemb[(1 * 5 + a1) * HD + hd]
                     + bemb[(2 * 5 + a2) * HD + hd];
      const float xj = hbuf[row * HD + hd] + em;   // message source + edge emb
      const float xi = hbuf[col * HD + hd];        // message target
      const float v  = att[h * HD + d] * xi + att[h * HD + DD + d] * xj;
      if (h == 0) p0 += v; else p1 += v;
    }
  }
#pragma unroll
  for (int off = 16; off > 0; off >>= 1) {
    p0 += __shfl_xor(p0, off, 32);
    p1 += __shfl_xor(p1, off, 32);
  }
  if (lane == 0) {
    const float v0 = (p0 >= 0.f) ? p0 : NEG_SLOPE * p0;   // leaky_relu
    const float v1 = (p1 >= 0.f) ? p1 : NEG_SLOPE * p1;
    alpha[e * 2 + 0] = v0;
    alpha[e * 2 + 1] = v1;
    atomicMaxF(&m[row * 2 + 0], v0);
    atomicMaxF(&m[row * 2 + 1], v1);
  }
}

// ---------------------------------------------------------------------------
// Kernel 3: ea = exp(alpha - m[row]); segment-sum into s. Thread per (e,h).
// alpha is overwritten in place with ea (re-written each launch by kernel 2).
// ---------------------------------------------------------------------------
__global__ __launch_bounds__(256)
void edge_exp(const int* __restrict__ eidx, float* __restrict__ alpha,
              const float* __restrict__ m, float* __restrict__ s) {
  const int idx = blockIdx.x * 256 + threadIdx.x;
  if (idx >= EE * HH) return;
  const int e   = idx >> 1;
  const int h   = idx & 1;
  const int row = eidx[e];
  const float ea = expf(alpha[idx] - m[row * 2 + h]);
  alpha[idx] = ea;
  atomicAdd(&s[row * 2 + h], ea);
}

// ---------------------------------------------------------------------------
// Kernel 4: normalized weighted messages, scatter-add into out[col], fused
// mean-over-heads (x0.5). One wave per edge, lane owns 2 feature dims.
// ---------------------------------------------------------------------------
__global__ __launch_bounds__(256)
void edge_scatter(const float* __restrict__ hbuf, const int* __restrict__ eidx,
                  const int* __restrict__ eattr, const float* __restrict__ bemb,
                  const float* __restrict__ ea, const float* __restrict__ s,
                  float* __restrict__ out) {
  const int lane = threadIdx.x & 31;
  const int e    = blockIdx.x * 8 + (threadIdx.x >> 5);
  if (e >= EE) return;
  const int row = eidx[e];
  const int col = eidx[EE + e];
  const int a0  = eattr[e * 3 + 0];
  const int a1  = eattr[e * 3 + 1];
  const int a2  = eattr[e * 3 + 2];
  const float w0 = ea[e * 2 + 0] / (s[row * 2 + 0] + 1e-16f);
  const float w1 = ea[e * 2 + 1] / (s[row * 2 + 1] + 1e-16f);

#pragma unroll
  for (int dd = 0; dd < 2; ++dd) {
    const int d = lane * 2 + dd;
    float acc;
    {   // head 0
      const int hd = d;
      const float em = bemb[a0 * HD + hd] + bemb[(5 + a1) * HD + hd]
                     + bemb[(10 + a2) * HD + hd];
      acc = w0 * (hbuf[row * HD + hd] + em);
    }
    {   // head 1
      const int hd = DD + d;
      const float em = bemb[a0 * HD + hd] + bemb[(5 + a1) * HD + hd]
                     + bemb[(10 + a2) * HD + hd];
      acc += w1 * (hbuf[row * HD + hd] + em);
    }
    atomicAdd(&out[col * DD + d], acc * 0.5f);   // mean over 2 heads
  }
}

// ---------------------------------------------------------------------------
extern "C" void kernel_launch(void* const* d_in, const int* in_sizes, int n_in,
                              void* d_out, int out_size, void* d_ws, size_t ws_size,
                              hipStream_t stream) {
  const float* x     = (const float*)d_in[0];  // [N, 64]
  const int*   eidx  = (const int*)  d_in[1];  // [2, E]
  const int*   eattr = (const int*)  d_in[2];  // [E, 3]
  const float* W     = (const float*)d_in[3];  // [64, 128]
  const float* bW    = (const float*)d_in[4];  // [128]
  const float* att   = (const float*)d_in[5];  // [1, 2, 128]
  const float* bias  = (const float*)d_in[6];  // [64]
  const float* bemb  = (const float*)d_in[7];  // [3, 5, 128]
  float*       out   = (float*)d_out;          // [N, 64]

  // Workspace layout (floats): hbuf | alpha/ea | m | s  (~39.2 MB total)
  float* hbuf  = (float*)d_ws;             // N*128
  float* alpha = hbuf  + (size_t)NN * HD;  // E*2
  float* m     = alpha + (size_t)EE * HH;  // N*2
  float* s     = m     + (size_t)NN * HH;  // N*2

  // K0: GEMM — 3125*8 = 25000 waves, 4 waves/block -> 6250 blocks
  gemm_xw<<<6250, 128, 0, stream>>>(x, W, bW, hbuf);
  // K1: init out/m/s — max range N*64 = 3.2M threads
  init_buffers<<<(NN * DD + 255) / 256, 256, 0, stream>>>(out, bias, m, s);
  // K2: logits + segment max — 8 edges per block
  edge_alpha<<<(EE + 7) / 8, 256, 0, stream>>>(hbuf, eidx, eattr, att, bemb,
                                               alpha, m);
  // K3: exp + segment sum — thread per (edge, head)
  edge_exp<<<(EE * HH + 255) / 256, 256, 0, stream>>>(eidx, alpha, m, s);
  // K4: normalize + scatter-add messages
  edge_scatter<<<(EE + 7) / 8, 256, 0, stream>>>(hbuf, eidx, eattr, bemb,
                                                 alpha, s, out);
}